// VNCMDLayer_249108103958
// MI455X (gfx1250) — compile-verified
//
#include <hip/hip_runtime.h>

// ---------------------------------------------------------------------------
// VNCMD layer, fused for MI455X (gfx1250, wave32).
//
// opedoub = (D^T D) of the second-difference matrix is pentadiagonal:
//   interior: A[i][i]=6, A[i][i±1]=-4, A[i][i±2]=1
//   boundary deltas only at: A[0][0]=2, A[0][1]=A[1][0]=-3,
//                            A[2047][2047]=2, A[2046][2047]=A[2047][2046]=-3
// CG matvec coef*(A@p) + (diagv+reg)*p runs as block-sparse
// V_WMMA_F32_16X16X4_F32: 16 CG columns per workgroup, 5 WMMAs per 16-row
// block (20-column band window aligned at mbase-2). Interior operands depend
// only on the lane -> one LDS table (ATBL) loaded ds_load_b64 just-in-time,
// so no persistent operand registers and no cndmask chains in the loop.
// p is padded with 2 zero guard rows on each side -> edge B-reads need no
// bounds checks. The 6 boundary matrix deltas are scalar fix-ups.
// x, r live in VGPRs in the WMMA C-tile lane layout; Ap is recomputed in a
// second WMMA pass after the p.Ap reduction to stay under the VGPR cap.
// ---------------------------------------------------------------------------

typedef float v2f __attribute__((ext_vector_type(2)));
typedef float v8f __attribute__((ext_vector_type(8)));

#define Nn 2048
#define Bb 8
#define Kk 4
#define NSYS 32
#define TWO_PI_F 6.28318530717958647692f
#define CG_REG_F 1e-6f
#define CG_TOL_F 1e-6f
#define CG_ITERS 30

// d_out layout (floats)
#define OUT_EIF 0
#define OUT_XM 65536
#define OUT_YM 131072
#define OUT_BSX 196608
#define OUT_BSY 212992
#define OUT_LAM 229376
#define OUT_SCAL 245760

// workspace layout (floats)
#define WS_SCAL 0
#define WS_U 64
#define WS_COS 16448
#define WS_SIN 81984
#define WS_RX 147520   // rhs_x, later aliased as xs output (disjoint block r/w)
#define WS_RY 213056   // rhs_y, later aliased as ys output
#define WS_DIF 278592  // deltaIF, later aliased as smooth output
#define WS_BSX 344128
#define WS_BSY 360512

// k4 dynamic LDS layout (floats):
//   [0 .. 32831]      rows -2..2049 x 16 cols  (guarded p array; Pv = smem+32)
//   [32832 .. 65599]  DV: diagv + reg, 2048 x 16
//   [65600 .. 66127]  RED: reduction scratch (512 + 16)
//   [66128 .. 66447]  ATBL: 5 slices x 32 lanes x v2f interior A operands
#define SM_PV 32
#define SM_DV 32832
#define SM_RED 65600
#define SM_ATBL 66128
#define SM_TOTAL 66448

// ---------------------------------------------------------------------------
// K1: hyperparameter refinement MLP -> scalars in ws
// scal[0]=new_alpha [1]=new_beta [2]=betathr [3]=1/new_alpha [4]=2/new_alpha [5]=2/betathr
// ---------------------------------------------------------------------------
__global__ void k1_hyper(const float* init_freqs, const float* alpha_p, const float* beta_p,
                         const int* iter_p, const float* fe_w1, const float* fe_b1,
                         const float* fe_w2, const float* fe_b2,
                         const float* pr_w1, const float* pr_b1,
                         const float* pr_w2, const float* pr_b2,
                         const float* pr_w3, const float* pr_b3,
                         const float* iter_w, float* scal) {
  __shared__ float red0[Bb], red1[Bb];
  const int t = threadIdx.x;
  const float alpha = alpha_p[0], beta = beta_p[0];
  if (t < Bb) {
    float avg = 0.f;
    for (int k = 0; k < Kk; ++k) avg += init_freqs[t * Kk + k];
    avg *= (1.f / Kk);
    float h1[32];
    for (int o = 0; o < 32; ++o) h1[o] = fmaxf(fe_w1[o] * avg + fe_b1[o], 0.f);
    float h2[16];
    for (int o = 0; o < 16; ++o) {
      float a = fe_b2[o];
      for (int j = 0; j < 32; ++j) a += fe_w2[o * 32 + j] * h1[j];
      h2[o] = fmaxf(a, 0.f);
    }
    float z[18];
    for (int o = 0; o < 16; ++o) z[o] = h2[o];
    z[16] = alpha; z[17] = beta;
    float z1[64];
    for (int o = 0; o < 64; ++o) {
      float a = pr_b1[o];
      for (int j = 0; j < 18; ++j) a += pr_w1[o * 18 + j] * z[j];
      z1[o] = fmaxf(a, 0.f);
    }
    float z2[32];
    for (int o = 0; o < 32; ++o) {
      float a = pr_b2[o];
      for (int j = 0; j < 64; ++j) a += pr_w2[o * 64 + j] * z1[j];
      z2[o] = fmaxf(a, 0.f);
    }
    float r0 = pr_b3[0], r1 = pr_b3[1];
    for (int j = 0; j < 32; ++j) { r0 += pr_w3[j] * z2[j]; r1 += pr_w3[32 + j] * z2[j]; }
    r0 = tanhf(r0); r1 = tanhf(r1);
    const float itf = (float)iter_p[0];
    const float fac = 1.f / (1.f + expf(-iter_w[0] * itf));
    red0[t] = r0 * fac * 0.1f;
    red1[t] = r1 * fac * 0.1f;
  }
  __syncthreads();
  if (t == 0) {
    float m0 = 0.f, m1 = 0.f;
    for (int b = 0; b < Bb; ++b) { m0 += red0[b]; m1 += red1[b]; }
    m0 *= (1.f / Bb); m1 *= (1.f / Bb);
    float na = fminf(fmaxf(alpha + m0 * alpha, 1e-6f), 0.01f);
    float nb = fminf(fmaxf(beta + m1 * beta, 1e-6f), 0.1f);
    const float itf = (float)iter_p[0];
    float bthr = fminf(exp10f(itf / 36.f - 10.f), nb);
    scal[0] = na; scal[1] = nb; scal[2] = bthr;
    scal[3] = 1.f / na; scal[4] = 2.f / na; scal[5] = 2.f / bthr;
  }
}

// ---------------------------------------------------------------------------
// K2: u = projec5(s - sum_x - sum_y - lamuda/new_alpha, var)   (per-b norm)
// ---------------------------------------------------------------------------
__global__ void k2_u(const float* s, const float* sum_x, const float* sum_y,
                     const float* lam, const float* var_p, const float* scal, float* u) {
  __shared__ float red[256];
  const int b = blockIdx.x, t = threadIdx.x;
  const float ia = scal[3];
  float tv[8]; float ssum = 0.f;
#pragma unroll
  for (int j = 0; j < 8; ++j) {
    const int gi = b * Nn + t * 8 + j;
    const float v = s[gi] - sum_x[gi] - sum_y[gi] - lam[gi] * ia;
    tv[j] = v; ssum += v * v;
  }
  red[t] = ssum; __syncthreads();
  for (int o = 128; o > 0; o >>= 1) { if (t < o) red[t] += red[t + o]; __syncthreads(); }
  const float n = sqrtf(red[0]);
  const float e = sqrtf((float)Nn * var_p[0]);
  const float sc = (n > e) ? (e / fmaxf(n, 1e-30f)) : 1.f;
#pragma unroll
  for (int j = 0; j < 8; ++j) u[b * Nn + t * 8 + j] = tv[j] * sc;
}

// ---------------------------------------------------------------------------
// K3: per (b,k): phase = 2pi*cumtrapz(eIF)  (workgroup scan), cos/sin, rhs
// ---------------------------------------------------------------------------
__global__ void k3_prep(const float* s, const float* eIF, const float* xm, const float* ym,
                        const float* sum_x, const float* sum_y, const float* lam,
                        const float* fs, const float* scal, const float* u,
                        float* cosw, float* sinw, float* rhsx, float* rhsy) {
  __shared__ float st[256];
  const int sys = blockIdx.x, b = sys >> 2, t = threadIdx.x;
  const float dx = 1.f / fs[0];
  const float ia = scal[3];
  const float* E = eIF + sys * Nn;
  float av[8]; float tot = 0.f;
#pragma unroll
  for (int j = 0; j < 8; ++j) {
    const int i = t * 8 + j;
    const float a = (i < Nn - 1) ? (E[i] + E[i + 1]) * 0.5f * dx : 0.f;
    av[j] = a; tot += a;
  }
  st[t] = tot; __syncthreads();
  for (int off = 1; off < 256; off <<= 1) {
    const float v = st[t];
    const float w = (t >= off) ? st[t - off] : 0.f;
    __syncthreads();
    st[t] = v + w;
    __syncthreads();
  }
  float run = (t > 0) ? st[t - 1] : 0.f;
#pragma unroll
  for (int j = 0; j < 8; ++j) {
    const int i = t * 8 + j;
    const int gi = sys * Nn + i, bi = b * Nn + i;
    const float ph = TWO_PI_F * run;
    float sv, cv; sincosf(ph, &sv, &cv);
    cosw[gi] = cv; sinw[gi] = sv;
    const float resid = s[bi] - (sum_x[bi] - xm[gi] * cv) - (sum_y[bi] - ym[gi] * sv)
                        - u[bi] - lam[bi] * ia;
    rhsx[gi] = cv * resid;
    rhsy[gi] = sv * resid;
    run += av[j];
  }
}

// ---------------------------------------------------------------------------
// K4: fused 30-iteration CG, 16 columns per workgroup, WMMA banded matvec.
// ---------------------------------------------------------------------------
__device__ __forceinline__ float reduce_col(float v, float* RED, int tid, int wave, int lane, int c) {
  v += __shfl_xor(v, 16, 32);          // fold lane L with L^16 -> per-wave column sum
  if (lane < 16) RED[wave * 16 + lane] = v;
  __syncthreads();
  if (tid < 16) {
    float s = 0.f;
#pragma unroll
    for (int w = 0; w < 32; ++w) s += RED[w * 16 + tid];
    RED[512 + tid] = s;
  }
  __syncthreads();
  const float out = RED[512 + c];
  __syncthreads();
  return out;
}

__global__ void __launch_bounds__(1024) k4_cg(int mode, const float* scal,
    const float* cosw, const float* sinw, const float* rhsx, const float* rhsy,
    const float* xm, const float* ym, const float* dif,
    float* xs, float* ys, float* sm) {
  extern __shared__ float smem[];
  float* Pv   = smem + SM_PV;          // rows -2..2049 valid (guards zeroed)
  float* DV   = smem + SM_DV;
  float* RED  = smem + SM_RED;
  float* ATBL = smem + SM_ATBL;

  const int tid = threadIdx.x;
  const int wave = tid >> 5;
  const int lane = tid & 31;
  const int c = lane & 15;                  // column within workgroup / WMMA N index
  const int hi = (lane >> 4) & 1;
  const int rowoff = hi * 8;                // C/D tile: lanes 16-31 hold rows M+8
  const int koff = hi * 2;                  // A/B tile: lanes 16-31 hold K+2
  const int wave4 = wave * 4;
  const int lane2 = lane * 2;

  const int col_id = blockIdx.x * 16 + c;
  int sys; const float* rhsp; const float* x0p; const float* trig; float* outp; float coef;
  if (mode == 0) {
    const int isy = col_id >= 32;
    sys  = col_id & 31;
    rhsp = isy ? rhsy : rhsx;
    x0p  = isy ? ym : xm;
    trig = isy ? sinw : cosw;
    outp = isy ? ys : xs;
    coef = scal[4];
  } else {
    sys = col_id; rhsp = dif; x0p = dif; trig = dif; outp = sm; coef = scal[5];
  }

  // One-time LDS setup: zero guard rows; build the interior A-operand table.
  // Interior band g(j-i): g(0)=6, g(+-1)=-4, g(+-2)=1, else 0.
  // Slice t covers band cols mbase-2+4t .. +3; lane holds A row M=c, K=koff{,+1}.
  if (tid < 32) {
    smem[tid] = 0.f;                 // guard rows -2,-1
    smem[SM_PV + Nn * 16 + tid] = 0.f;  // guard rows 2048,2049
#pragma unroll
    for (int t5 = 0; t5 < 5; ++t5) {
      const int jrel = -2 + 4 * t5;
      const int d0 = jrel + koff - c;
      const int d1 = jrel + koff + 1 - c;
      const int a0 = d0 < 0 ? -d0 : d0;
      const int a1 = d1 < 0 ? -d1 : d1;
      ATBL[t5 * 64 + lane2]     = (a0 == 0) ? 6.f : ((a0 == 1) ? -4.f : ((a0 == 2) ? 1.f : 0.f));
      ATBL[t5 * 64 + lane2 + 1] = (a1 == 0) ? 6.f : ((a1 == 1) ? -4.f : ((a1 == 2) ? 1.f : 0.f));
    }
  }

  // Load diagv(+reg) and the initial multiply vector (x0) into LDS.
#pragma unroll
  for (int mloc = 0; mloc < 4; ++mloc) {
    const int lib = ((wave4 + mloc) * 16 + rowoff) * 16 + c;
    const int gib = sys * Nn + (wave4 + mloc) * 16 + rowoff;
#pragma unroll
    for (int v = 0; v < 8; ++v) {
      if (mode == 0) {
        const float tg = trig[gib + v];
        DV[lib + v * 16] = tg * tg + CG_REG_F;
        Pv[lib + v * 16] = x0p[gib + v];
      } else {
        DV[lib + v * 16] = 1.f + CG_REG_F;
        Pv[lib + v * 16] = 0.f;
      }
    }
  }
  __syncthreads();

  // banded A@P tile for row block mloc of this wave: 5 WMMAs, operands from LDS,
  // guard rows make edge reads branch-free.
  auto acc_tile = [&](int mloc) -> v8f {
    const int base = ((wave4 + mloc) * 16 - 2 + koff) * 16 + c;
    v8f acc = {};
#pragma unroll
    for (int t5 = 0; t5 < 5; ++t5) {
      v2f av = *(const v2f*)(ATBL + t5 * 64 + lane2);
      v2f bb;
      bb.x = Pv[base + t5 * 64];
      bb.y = Pv[base + t5 * 64 + 16];
      acc = __builtin_amdgcn_wmma_f32_16x16x4_f32(false, av, false, bb,
                                                  (short)0, acc, false, false);
    }
    return acc;
  };

  // Boundary deltas vs interior stencil (rows 0,1,2046,2047 only).
  auto bfix = [&](int mloc, int v) -> float {
    float fx = 0.f;
    if (mloc == 0 && v == 0 && wave == 0 && hi == 0) fx = -4.f * Pv[c] + Pv[16 + c];
    if (mloc == 0 && v == 1 && wave == 0 && hi == 0) fx = Pv[c];
    if (mloc == 3 && v == 6 && wave == 31 && hi == 1) fx = Pv[2047 * 16 + c];
    if (mloc == 3 && v == 7 && wave == 31 && hi == 1) fx = Pv[2046 * 16 + c] - 4.f * Pv[2047 * 16 + c];
    return fx;
  };

  float x[32], r[32];

  // r0 = rhs - mv(x0);  x = x0;  p = r0
  float rs_part = 0.f;
#pragma unroll
  for (int mloc = 0; mloc < 4; ++mloc) {
    v8f acc = acc_tile(mloc);
    const int lib = ((wave4 + mloc) * 16 + rowoff) * 16 + c;
    const int gib = sys * Nn + (wave4 + mloc) * 16 + rowoff;
#pragma unroll
    for (int v = 0; v < 8; ++v) {
      const int li = lib + v * 16;
      const float pv = Pv[li];
      const float apv = coef * (acc[v] + bfix(mloc, v)) + DV[li] * pv;
      const int idx = mloc * 8 + v;
      x[idx] = pv;
      r[idx] = rhsp[gib + v] - apv;
      rs_part += r[idx] * r[idx];
    }
  }
  __syncthreads();   // all matvec reads of x0 done before overwriting P
#pragma unroll
  for (int mloc = 0; mloc < 4; ++mloc) {
    const int lib = ((wave4 + mloc) * 16 + rowoff) * 16 + c;
#pragma unroll
    for (int v = 0; v < 8; ++v) Pv[lib + v * 16] = r[mloc * 8 + v];
  }
  float rsold = reduce_col(rs_part, RED, tid, wave, lane, c);
  bool done = false;

  for (int it = 0; it < CG_ITERS; ++it) {
    __syncthreads();
    // pass 1: pp = p . Ap   (Ap not stored)
    float pp = 0.f;
#pragma unroll
    for (int mloc = 0; mloc < 4; ++mloc) {
      v8f acc = acc_tile(mloc);
      const int lib = ((wave4 + mloc) * 16 + rowoff) * 16 + c;
#pragma unroll
      for (int v = 0; v < 8; ++v) {
        const int li = lib + v * 16;
        const float pv = Pv[li];
        const float apv = coef * (acc[v] + bfix(mloc, v)) + DV[li] * pv;
        pp += pv * apv;
      }
    }
    const float pAp = reduce_col(pp, RED, tid, wave, lane, c);
    const float a = rsold / (pAp + 1e-12f);
    // pass 2: recompute Ap, update x, r  (frozen when done)
    float rp = 0.f;
#pragma unroll
    for (int mloc = 0; mloc < 4; ++mloc) {
      v8f acc = acc_tile(mloc);
      const int lib = ((wave4 + mloc) * 16 + rowoff) * 16 + c;
#pragma unroll
      for (int v = 0; v < 8; ++v) {
        const int li = lib + v * 16;
        const float pv = Pv[li];
        const float apv = coef * (acc[v] + bfix(mloc, v)) + DV[li] * pv;
        const int idx = mloc * 8 + v;
        if (!done) {
          x[idx] += a * pv;
          r[idx] -= a * apv;
        }
        rp += r[idx] * r[idx];
      }
    }
    const float rsnew = reduce_col(rp, RED, tid, wave, lane, c);
    const bool done1 = done || (sqrtf(rsnew) < CG_TOL_F);
    const float bta = rsnew / (rsold + 1e-12f);
    if (!done1) {
#pragma unroll
      for (int mloc = 0; mloc < 4; ++mloc) {
        const int lib = ((wave4 + mloc) * 16 + rowoff) * 16 + c;
#pragma unroll
        for (int v = 0; v < 8; ++v) {
          const int li = lib + v * 16;
          Pv[li] = r[mloc * 8 + v] + bta * Pv[li];
        }
      }
      rsold = rsnew;
    }
    done = done1;
  }

#pragma unroll
  for (int mloc = 0; mloc < 4; ++mloc) {
    const int gib = sys * Nn + (wave4 + mloc) * 16 + rowoff;
#pragma unroll
    for (int v = 0; v < 8; ++v) outp[gib + v] = x[mloc * 8 + v];
  }
}

// ---------------------------------------------------------------------------
// K5: deltaIF = (xs*ybar - ys*xbar) / ((xs^2+ys^2+1e-12)*2pi)   (differ5)
// ---------------------------------------------------------------------------
__global__ void k5_delta(const float* xs, const float* ys, const float* fs, float* dif) {
  const int sys = blockIdx.x, t = threadIdx.x;
  const float dx = 1.f / fs[0];
  const float* X = xs + sys * Nn;
  const float* Y = ys + sys * Nn;
#pragma unroll
  for (int j = 0; j < 8; ++j) {
    const int i = t * 8 + j;
    float xb, yb;
    if (i == 0) { xb = (X[1] - X[0]) / dx; yb = (Y[1] - Y[0]) / dx; }
    else if (i == Nn - 1) { xb = (X[Nn - 1] - X[Nn - 2]) / dx; yb = (Y[Nn - 1] - Y[Nn - 2]) / dx; }
    else { xb = (X[i + 1] - X[i - 1]) / (2.f * dx); yb = (Y[i + 1] - Y[i - 1]) / (2.f * dx); }
    const float xv = X[i], yv = Y[i];
    const float den = xv * xv + yv * yv + 1e-12f;
    dif[sys * Nn + i] = (xv * yb - yv * xb) / (den * TWO_PI_F);
  }
}

// ---------------------------------------------------------------------------
// K6: per-b block: masked outputs, new_phase scan per k, bsx/bsy (no atomics)
// ---------------------------------------------------------------------------
__global__ void k6_out(const float* eIF, const float* xm, const float* ym,
                       const int* mask, const float* fs,
                       const float* smw, const float* xsw, const float* ysw,
                       float* out, float* bsx, float* bsy) {
  __shared__ float neif[Nn];
  __shared__ float st[256];
  const int b = blockIdx.x, t = threadIdx.x;
  const float dx = 1.f / fs[0];
  float bx[8], by[8];
#pragma unroll
  for (int j = 0; j < 8; ++j) { bx[j] = 0.f; by[j] = 0.f; }

  for (int k = 0; k < Kk; ++k) {
    __syncthreads();
    const int sys = b * Kk + k;
    const bool act = (mask[sys] != 0);
    float nx[8], ny[8];
#pragma unroll
    for (int j = 0; j < 8; ++j) {
      const int i = t * 8 + j;
      const int gi = sys * Nn + i;
      const float e0 = eIF[gi];
      const float ne = act ? (e0 - 0.5f * smw[gi]) : e0;
      out[OUT_EIF + gi] = ne;
      const float vx = act ? xsw[gi] : xm[gi];
      const float vy = act ? ysw[gi] : ym[gi];
      nx[j] = vx; ny[j] = vy;
      out[OUT_XM + gi] = vx;
      out[OUT_YM + gi] = vy;
      neif[i] = ne;
    }
    __syncthreads();
    float av[8]; float tot = 0.f;
#pragma unroll
    for (int j = 0; j < 8; ++j) {
      const int i = t * 8 + j;
      const float a = (i < Nn - 1) ? (neif[i] + neif[i + 1]) * 0.5f * dx : 0.f;
      av[j] = a; tot += a;
    }
    st[t] = tot; __syncthreads();
    for (int off = 1; off < 256; off <<= 1) {
      const float v = st[t];
      const float w = (t >= off) ? st[t - off] : 0.f;
      __syncthreads();
      st[t] = v + w;
      __syncthreads();
    }
    float run = (t > 0) ? st[t - 1] : 0.f;
    const float mf = act ? 1.f : 0.f;
#pragma unroll
    for (int j = 0; j < 8; ++j) {
      const float ph = TWO_PI_F * run;
      float sv, cv; sincosf(ph, &sv, &cv);
      bx[j] += mf * nx[j] * cv;
      by[j] += mf * ny[j] * sv;
      run += av[j];
    }
  }
#pragma unroll
  for (int j = 0; j < 8; ++j) {
    const int gi = b * Nn + t * 8 + j;
    bsx[gi] = bx[j];
    bsy[gi] = by[j];
  }
}

// ---------------------------------------------------------------------------
// K7: new_lamuda = lamuda + new_alpha*(u + bsx + bsy - s); copy bsx/bsy/scalars
// ---------------------------------------------------------------------------
__global__ void k7_fin(const float* s, const float* lam, const float* u,
                       const float* bsx, const float* bsy, const float* scal, float* out) {
  const int b = blockIdx.x, t = threadIdx.x;
  const float na = scal[0];
#pragma unroll
  for (int j = 0; j < 8; ++j) {
    const int gi = b * Nn + t * 8 + j;
    const float bx = bsx[gi], by = bsy[gi];
    out[OUT_BSX + gi] = bx;
    out[OUT_BSY + gi] = by;
    out[OUT_LAM + gi] = lam[gi] + na * (u[gi] + bx + by - s[gi]);
  }
  if (b == 0 && t == 0) { out[OUT_SCAL] = scal[0]; out[OUT_SCAL + 1] = scal[1]; }
}

// ---------------------------------------------------------------------------
extern "C" void kernel_launch(void* const* d_in, const int* in_sizes, int n_in,
                              void* d_out, int out_size, void* d_ws, size_t ws_size,
                              hipStream_t stream) {
  (void)in_sizes; (void)n_in; (void)out_size; (void)ws_size;
  const float* s          = (const float*)d_in[0];
  const float* eIF        = (const float*)d_in[1];
  const float* xm         = (const float*)d_in[2];
  const float* ym         = (const float*)d_in[3];
  const float* sum_x      = (const float*)d_in[4];
  const float* sum_y      = (const float*)d_in[5];
  const float* lam        = (const float*)d_in[6];
  const float* init_freqs = (const float*)d_in[7];
  const int*   mode_mask  = (const int*)d_in[8];
  const float* alpha      = (const float*)d_in[9];
  const float* beta       = (const float*)d_in[10];
  const float* var        = (const float*)d_in[11];
  const float* fs         = (const float*)d_in[12];
  const int*   iteration  = (const int*)d_in[13];
  const float* fe_w1      = (const float*)d_in[14];
  const float* fe_b1      = (const float*)d_in[15];
  const float* fe_w2      = (const float*)d_in[16];
  const float* fe_b2      = (const float*)d_in[17];
  const float* pr_w1      = (const float*)d_in[18];
  const float* pr_b1      = (const float*)d_in[19];
  const float* pr_w2      = (const float*)d_in[20];
  const float* pr_b2      = (const float*)d_in[21];
  const float* pr_w3      = (const float*)d_in[22];
  const float* pr_b3      = (const float*)d_in[23];
  const float* iter_w     = (const float*)d_in[24];

  float* out  = (float*)d_out;
  float* ws   = (float*)d_ws;
  float* scal = ws + WS_SCAL;
  float* u    = ws + WS_U;
  float* cosw = ws + WS_COS;
  float* sinw = ws + WS_SIN;
  float* rx   = ws + WS_RX;   // rhs_x then xs
  float* ry   = ws + WS_RY;   // rhs_y then ys
  float* dif  = ws + WS_DIF;  // deltaIF then smooth
  float* bsx  = ws + WS_BSX;
  float* bsy  = ws + WS_BSY;

  k1_hyper<<<1, 32, 0, stream>>>(init_freqs, alpha, beta, iteration,
                                 fe_w1, fe_b1, fe_w2, fe_b2,
                                 pr_w1, pr_b1, pr_w2, pr_b2, pr_w3, pr_b3,
                                 iter_w, scal);
  k2_u<<<Bb, 256, 0, stream>>>(s, sum_x, sum_y, lam, var, scal, u);
  k3_prep<<<NSYS, 256, 0, stream>>>(s, eIF, xm, ym, sum_x, sum_y, lam, fs, scal, u,
                                    cosw, sinw, rx, ry);

  const size_t smem = (size_t)SM_TOTAL * sizeof(float);  // 265792 B < 320KB WGP LDS
  // xs/ys solves: 64 columns -> 4 workgroups of 16 columns
  k4_cg<<<4, 1024, smem, stream>>>(0, scal, cosw, sinw, rx, ry, xm, ym, dif, rx, ry, dif);
  k5_delta<<<NSYS, 256, 0, stream>>>(rx, ry, fs, dif);
  // smooth solves: 32 columns -> 2 workgroups
  k4_cg<<<2, 1024, smem, stream>>>(1, scal, cosw, sinw, rx, ry, xm, ym, dif, rx, ry, dif);

  k6_out<<<Bb, 256, 0, stream>>>(eIF, xm, ym, mode_mask, fs, dif, rx, ry, out, bsx, bsy);
  k7_fin<<<Bb, 256, 0, stream>>>(s, lam, u, bsx, bsy, scal, out);
}